// GeometryTransformer_22943715295494
// MI455X (gfx1250) — compile-verified
//
#include <hip/hip_runtime.h>
#include <math.h>

// Problem constants (fixed by the reference harness).
#define B_  2
#define V_  6
#define C_  128
#define HF  135
#define WF  240
#define BEVH 200
#define BEVW 200
#define NPIX (BEVH * BEVW)       // 40000 points per camera
#define NBV  (B_ * V_)           // 12 cameras
#define PLANE (HF * WF)          // 32400 floats per channel plane

typedef float v2f __attribute__((ext_vector_type(2)));
typedef float v8f __attribute__((ext_vector_type(8)));

// ---------------------------------------------------------------------------
// Kernel 1: homography projection as an f32 WMMA GEMM.
//   H(3x3) x [x; y; 1](3 x N)  ->  (u,v,w) per point  ->  fx,fy into ws.
// One wave handles 16 points with a single V_WMMA_F32_16X16X4_F32:
//   A = H padded to 16x4 (rows 3..15 zero, K=3 unused),
//   B = ground coords padded to 4x16,
//   D rows 0..2 = u,v,w for the 16 points (columns).
// 2500 waves per camera (40000/16), 30000 waves total.
// ---------------------------------------------------------------------------
__global__ __launch_bounds__(256)
void bev_proj_wmma(const float* __restrict__ intr,   // (B,V,3,3)
                   const float* __restrict__ extr,   // (B,V,4,4)
                   const int*   __restrict__ Himg,
                   const int*   __restrict__ Wimg,
                   float*       __restrict__ fxfy)   // (NBV*NPIX) float2
{
    const int tid  = threadIdx.x;
    const int lane = tid & 31;
    const int n    = lane & 15;                       // point-within-wave
    const int wave = blockIdx.x * (blockDim.x >> 5) + (tid >> 5);
    const int bv   = wave / (NPIX / 16);              // 2500 waves per camera
    const int pl   = (wave % (NPIX / 16)) * 16 + n;   // local point index
    const int px   = pl % BEVW;
    const int py   = pl / BEVW;

    // Ground-plane grid (bounds [-50,50], res 0.5): g = (x, y, 1).
    const float gx = -50.0f + ((float)px + 0.5f) * 0.5f;
    const float gy = -50.0f + ((float)py + 0.5f) * 0.5f;

    // H = K @ [R[:,0], R[:,1], t]  (each lane computes the 3x3, it's 27 FMAs).
    const float* K = intr + bv * 9;
    const float* E = extr + bv * 16;
    float Hm[9];
#pragma unroll
    for (int i = 0; i < 3; ++i) {
#pragma unroll
        for (int j = 0; j < 3; ++j) {
            const int col = (j == 2) ? 3 : j;         // cols 0,1 of R, then t
            float s = 0.0f;
#pragma unroll
            for (int k = 0; k < 3; ++k) s = fmaf(K[i * 3 + k], E[k * 4 + col], s);
            Hm[i * 3 + j] = s;
        }
    }

    // A 16x4 f32 layout: lanes 0-15 -> M=lane, VGPR0=K0, VGPR1=K1;
    //                    lanes 16-31 -> M=lane-16, VGPR0=K2, VGPR1=K3.
    // B 4x16 layout mirrors the K split across lane halves.
    v2f a, b;
    if (lane < 16) {
        a.x = (lane < 3) ? Hm[lane * 3 + 0] : 0.0f;   // K=0 column of H
        a.y = (lane < 3) ? Hm[lane * 3 + 1] : 0.0f;   // K=1
        b.x = gx;                                     // B row 0
        b.y = gy;                                     // B row 1
    } else {
        const int m = lane - 16;
        a.x = (m < 3) ? Hm[m * 3 + 2] : 0.0f;         // K=2
        a.y = 0.0f;                                   // K=3 (unused)
        b.x = 1.0f;                                   // B row 2 (homogeneous 1)
        b.y = 0.0f;                                   // B row 3 (unused)
    }

    v8f acc = {};
    // D = A x B + 0  ->  v_wmma_f32_16x16x4_f32
    acc = __builtin_amdgcn_wmma_f32_16x16x4_f32(
        /*neg_a=*/false, a, /*neg_b=*/false, b,
        /*c_mod=*/(short)0, acc, /*reuse_a=*/false, /*reuse_b=*/false);

    // D row r -> VGPR r on lanes 0-15: u=acc[0], v=acc[1], w=acc[2].
    if (lane < 16) {
        const float u = acc[0], v = acc[1], w = acc[2];
        const float wsafe = (fabsf(w) < 1e-6f) ? 1.0f : w;
        const float inv   = 1.0f / wsafe;
        const float sx = (float)WF / (float)(*Wimg);
        const float sy = (float)HF / (float)(*Himg);
        float2 r;
        r.x = u * inv * sx;
        r.y = v * inv * sy;
        ((float2*)fxfy)[(size_t)bv * NPIX + pl] = r;
    }
}

// ---------------------------------------------------------------------------
// Kernel 2: bilinear gather. One thread per BEV pixel, loop over 128 channels.
// Corner validity folded into weights; indices clamped so all 4 loads are
// unconditional and in-bounds. Fast path: if all four weights are zero
// (pixel outside this camera's frustum / image — the common case for
// surround-view), emit only the 128 zero stores and skip all 512 loads.
// Output stores are non-temporal (streaming 246 MB; keep L2 for the
// gather reuse).
// ---------------------------------------------------------------------------
__global__ __launch_bounds__(256)
void bev_sample(const float* __restrict__ feats,     // (NBV, C, HF, WF)
                const float* __restrict__ fxfy,      // (NBV*NPIX) float2
                float*       __restrict__ out)       // (NBV, C, BEVH, BEVW)
{
    const int bv = blockIdx.x;
    const int p  = blockIdx.y * blockDim.x + threadIdx.x;
    if (p >= NPIX) return;

    const float2 c = ((const float2*)fxfy)[(size_t)bv * NPIX + p];
    const float fx = c.x, fy = c.y;

    const float x0f = floorf(fx);
    const float y0f = floorf(fy);
    const int x0 = (int)x0f;
    const int y0 = (int)y0f;
    const float wx1 = fx - x0f, wy1 = fy - y0f;
    const float wx0 = 1.0f - wx1, wy0 = 1.0f - wy1;

    const bool vx0 = (x0 >= 0) & (x0 < WF);
    const bool vx1 = (x0 + 1 >= 0) & (x0 + 1 < WF);
    const bool vy0 = (y0 >= 0) & (y0 < HF);
    const bool vy1 = (y0 + 1 >= 0) & (y0 + 1 < HF);

    const float w00 = wx0 * wy0 * (float)(vx0 & vy0);
    const float w01 = wx1 * wy0 * (float)(vx1 & vy0);
    const float w10 = wx0 * wy1 * (float)(vx0 & vy1);
    const float w11 = wx1 * wy1 * (float)(vx1 & vy1);

    float* __restrict__ ob = out + (size_t)bv * C_ * NPIX + p;

    // All-corners-dead fast path: weights are non-negative, so a zero sum
    // means every contribution is exactly zero. Spatially coherent branch —
    // whole waves take one side almost everywhere.
    if (w00 + w01 + w10 + w11 == 0.0f) {
#pragma unroll 8
        for (int ch = 0; ch < C_; ++ch) {
            __builtin_nontemporal_store(0.0f, ob);
            ob += NPIX;
        }
        return;
    }

    const int xc0 = min(max(x0, 0), WF - 1);
    const int xc1 = min(max(x0 + 1, 0), WF - 1);
    const int yc0 = min(max(y0, 0), HF - 1);
    const int yc1 = min(max(y0 + 1, 0), HF - 1);

    const int i00 = yc0 * WF + xc0;
    const int i01 = yc0 * WF + xc1;
    const int i10 = yc1 * WF + xc0;
    const int i11 = yc1 * WF + xc1;

    const float* __restrict__ fb = feats + (size_t)bv * C_ * PLANE;

#pragma unroll 4
    for (int ch = 0; ch < C_; ++ch) {
        const float s = w00 * fb[i00] + w01 * fb[i01] +
                        w10 * fb[i10] + w11 * fb[i11];
        __builtin_nontemporal_store(s, ob);
        fb += PLANE;
        ob += NPIX;
    }
}

// ---------------------------------------------------------------------------
extern "C" void kernel_launch(void* const* d_in, const int* in_sizes, int n_in,
                              void* d_out, int out_size, void* d_ws, size_t ws_size,
                              hipStream_t stream)
{
    const float* feats = (const float*)d_in[0];  // (2,6,128,135,240) f32
    const float* intr  = (const float*)d_in[1];  // (2,6,3,3) f32
    const float* extr  = (const float*)d_in[2];  // (2,6,4,4) f32
    const int*   Himg  = (const int*)d_in[3];    // scalar 1080
    const int*   Wimg  = (const int*)d_in[4];    // scalar 1920
    float*       out   = (float*)d_out;          // (2,6,128,200,200) f32
    float*       fxfy  = (float*)d_ws;           // 480000 float2 = 3.84 MB

    // Stage 1: 30000 waves, 16 points each -> 3750 blocks of 256 threads.
    const int waves_total = NBV * (NPIX / 16);
    bev_proj_wmma<<<waves_total / 8, 256, 0, stream>>>(intr, extr, Himg, Wimg, fxfy);

    // Stage 2: one block column per camera, 157 tiles of 256 pixels.
    dim3 grid(NBV, (NPIX + 255) / 256);
    bev_sample<<<grid, 256, 0, stream>>>(feats, fxfy, out);
}